// SimpleAttention_50010599194765
// MI455X (gfx1250) — compile-verified
//
#include <hip/hip_runtime.h>
#include <hip/hip_bf16.h>

// ---------------------------------------------------------------------------
// Additive attention, fused, for MI455X (gfx1250, wave32).
//   K1: att_h = h @ W_h^T + b_h            (WMMA f32 16x16x4)
//   K2: scores -> masked softmax weights   (async global->LDS double-buffered
//       p tiles + v_tanh_f32 TRANS + shfl reduce)
//   K3: out = weight @ att_feats           (WMMA f32 16x16x4)
// Sizes: B=4, Lq=Lv=256, R=1024, A=512, D=1024.
// ---------------------------------------------------------------------------

typedef __attribute__((ext_vector_type(2))) float v2f;
typedef __attribute__((ext_vector_type(8))) float v8f;

#define BQ   4
#define LQ   256
#define LV   256
#define RNN  1024
#define AH   512
#define DD   1024

// ---------------------------------------------------------------------------
// K1: att_h[m, a] = sum_r h[m, r] * W_h[a, r] + b_h[a]
//     M = B*Lq = 1024, K = RNN = 1024, N = AH = 512, all f32.
// One wave per 16x16 output tile. A-frag: a[j] = h[m][k + 2*(lane/16) + j]
// (ISA 32-bit A 16x4 layout). B-frag: b[j] = B[2*(lane/16)+j][n] = W_h[n][k..]
// -> both are contiguous float2 loads.
// ---------------------------------------------------------------------------
__global__ __launch_bounds__(128) void k_atth(
    const float* __restrict__ h,     // [1024, 1024]
    const float* __restrict__ Wh,    // [512, 1024]
    const float* __restrict__ bh,    // [512]
    float* __restrict__ atth)        // [1024, 512]
{
    const int NT = AH / 16;                               // 32 tiles in N
    int wave = blockIdx.x * (blockDim.x >> 5) + (threadIdx.x >> 5);
    int lane = threadIdx.x & 31;
    int tm = wave / NT;
    int tn = wave % NT;

    int m    = tm * 16 + (lane & 15);
    int n    = tn * 16 + (lane & 15);
    int koff = (lane >> 4) * 2;

    const float* hrow = h  + m * RNN;
    const float* wrow = Wh + n * RNN;

    v8f acc = {};
    for (int k = 0; k < RNN; k += 4) {
        v2f a = *(const v2f*)(hrow + k + koff);
        v2f b = *(const v2f*)(wrow + k + koff);
        acc = __builtin_amdgcn_wmma_f32_16x16x4_f32(
            false, a, false, b, (short)0, acc, false, false);
    }

    // D layout: acc[i] -> row tm*16 + i + 8*(lane/16), col n
    int   rbase = tm * 16 + 8 * (lane >> 4);
    float bias  = bh[n];
#pragma unroll
    for (int i = 0; i < 8; ++i)
        atth[(rbase + i) * AH + n] = acc[i] + bias;
}

// ---------------------------------------------------------------------------
// K2: per (b,q) row (one wave each):
//   s[v]   = sum_a w_a[a] * tanh(p[b,v,a] + att_h[b,q,a]) + b_a
//   weight = m * exp(s - max) / sum(m * exp(s - max))   (softmax norm cancels)
//
// p[b] tiles (16 v-rows, 32 KB) are DMA'd into LDS with
// global_load_async_to_lds_b128 (ASYNCcnt), double-buffered, and shared by
// all 8 waves of the block (8x cut in vector-memory traffic). Compute path:
// native v_tanh_f32 TRANS op co-executing with the FMA reduce.
// Block = 256 threads = 8 waves = 8 q rows of one batch.
// ---------------------------------------------------------------------------
#define VT      16                    // v rows per LDS tile
#define TILE_F  (VT * AH)             // floats per tile (8192)
#define NTILES  (LV / VT)             // 16

__global__ __launch_bounds__(256) void k_scores(
    const float* __restrict__ atth,  // [B*Lq, AH]
    const float* __restrict__ p,     // [B, LV, AH]
    const int*   __restrict__ mask,  // [B, LV]
    const float* __restrict__ wa,    // [AH]
    const float* __restrict__ ba,    // [1]
    float* __restrict__ weight)      // [B, LQ, LV]
{
    __shared__ float smem[2 * TILE_F];                   // 64 KB

    int wave = threadIdx.x >> 5;
    int lane = threadIdx.x & 31;
    int b    = blockIdx.x >> 5;        // 32 q-tiles of 8 per batch
    int qt   = blockIdx.x & 31;
    int q    = qt * 8 + wave;
    int row  = b * LQ + q;

    float ah[16], wv[16];
#pragma unroll
    for (int i = 0; i < 16; ++i) {
        ah[i] = atth[row * AH + lane + 32 * i];
        wv[i] = wa[lane + 32 * i];
    }

    const float* pb   = p + (size_t)b * LV * AH;
    float        bias = ba[0];
    float        scv[8];                // lane owns v = 32*i + lane

    // LDS byte offset of smem (generic-pointer low 32 bits == LDS offset)
    uint32_t lds0 = (uint32_t)(size_t)&smem[0];

    // issue one tile's async DMA: 32 KB = 256 threads x 8 x b128
#define ISSUE_TILE(tile, buf)                                                  \
    do {                                                                       \
        const char* _src = (const char*)(pb + (size_t)(tile) * TILE_F);        \
        uint32_t    _dst = lds0 + (buf) * (TILE_F * 4);                        \
        _Pragma("unroll")                                                      \
        for (int _j = 0; _j < (TILE_F * 4) / (256 * 16); ++_j) {               \
            uint32_t _off = (threadIdx.x + 256 * _j) * 16;                     \
            uint64_t _ga  = (uint64_t)_src + _off;                             \
            uint32_t _la  = _dst + _off;                                       \
            asm volatile("global_load_async_to_lds_b128 %0, %1, off"           \
                         :: "v"(_la), "v"(_ga) : "memory");                    \
        }                                                                      \
    } while (0)

    ISSUE_TILE(0, 0);

    for (int tile = 0; tile < NTILES; ++tile) {
        int buf = tile & 1;
        asm volatile("s_wait_asynccnt 0x0" ::: "memory");  // my tile loads done
        __syncthreads();                                   // everyone's done
        if (tile + 1 < NTILES) ISSUE_TILE(tile + 1, buf ^ 1);

        const float* ps = &smem[buf * TILE_F];
        for (int vl = 0; vl < VT; ++vl) {
            int   v = tile * VT + vl;
            float s = 0.f;
#pragma unroll
            for (int i = 0; i < 16; ++i) {
                float x = ps[vl * AH + lane + 32 * i] + ah[i];
                float t;
                asm("v_tanh_f32 %0, %1" : "=v"(t) : "v"(x)); // native TRANS op
                s = fmaf(wv[i], t, s);
            }
#pragma unroll
            for (int off = 16; off; off >>= 1)
                s += __shfl_xor(s, off, 32);
            if (lane == (v & 31)) scv[v >> 5] = s + bias;
        }
        __syncthreads();             // all waves finished this buffer
    }
#undef ISSUE_TILE

    // masked softmax over 256 scores (8 per lane), softmax norm cancels.
    float mx = -3.402823466e+38f;
#pragma unroll
    for (int i = 0; i < 8; ++i) mx = fmaxf(mx, scv[i]);
#pragma unroll
    for (int off = 16; off; off >>= 1)
        mx = fmaxf(mx, __shfl_xor(mx, off, 32));

    float e[8], msum = 0.f;
#pragma unroll
    for (int i = 0; i < 8; ++i) {
        float ex = __expf(scv[i] - mx);                    // v_exp_f32
        float mk = (float)mask[b * LV + 32 * i + lane];
        e[i]  = ex * mk;
        msum += e[i];
    }
#pragma unroll
    for (int off = 16; off; off >>= 1)
        msum += __shfl_xor(msum, off, 32);

    float inv = 1.0f / msum;
#pragma unroll
    for (int i = 0; i < 8; ++i)
        weight[(size_t)row * LV + 32 * i + lane] = e[i] * inv;
}

// ---------------------------------------------------------------------------
// K3: out[b] = weight[b] (256x256) @ att_feats[b] (256x1024), f32 WMMA.
// One wave per 16x16 tile, K-loop of 64 x (16x16x4) WMMAs.
// ---------------------------------------------------------------------------
__global__ __launch_bounds__(256) void k_out(
    const float* __restrict__ weight, // [B, LQ, LV]
    const float* __restrict__ vfeat,  // [B, LV, DD]
    float* __restrict__ out)          // [B, LQ, DD]
{
    const int MT = LQ / 16;                         // 16
    const int NT = DD / 16;                         // 64
    int wave = blockIdx.x * (blockDim.x >> 5) + (threadIdx.x >> 5);
    int lane = threadIdx.x & 31;

    int b  = wave / (MT * NT);
    int t  = wave % (MT * NT);
    int tm = t / NT;
    int tn = t % NT;

    const float* W = weight + (size_t)b * LQ * LV;
    const float* V = vfeat  + (size_t)b * LV * DD;

    int m    = tm * 16 + (lane & 15);
    int n    = tn * 16 + (lane & 15);
    int koff = (lane >> 4) * 2;

    v8f acc = {};
    for (int k = 0; k < LV; k += 4) {
        v2f a = *(const v2f*)(W + m * LV + k + koff);  // A[m][k..k+1]
        v2f bb;
        bb.x = V[(k + koff + 0) * DD + n];             // B[k][n] row-major V
        bb.y = V[(k + koff + 1) * DD + n];
        acc = __builtin_amdgcn_wmma_f32_16x16x4_f32(
            false, a, false, bb, (short)0, acc, false, false);
    }

    int rbase = tm * 16 + 8 * (lane >> 4);
#pragma unroll
    for (int i = 0; i < 8; ++i)
        out[(size_t)b * LQ * DD + (rbase + i) * DD + n] = acc[i];
}

// ---------------------------------------------------------------------------
extern "C" void kernel_launch(void* const* d_in, const int* in_sizes, int n_in,
                              void* d_out, int out_size, void* d_ws, size_t ws_size,
                              hipStream_t stream) {
    const float* h     = (const float*)d_in[0];   // [4,256,1024]
    const float* feats = (const float*)d_in[1];   // [4,256,1024]
    const float* p     = (const float*)d_in[2];   // [4,256,512]
    const int*   masks = (const int*)  d_in[3];   // [4,256]
    const float* Wh    = (const float*)d_in[4];   // [512,1024]
    const float* bh    = (const float*)d_in[5];   // [512]
    const float* wa    = (const float*)d_in[6];   // [512]
    const float* ba    = (const float*)d_in[7];   // [1]
    float*       outp  = (float*)d_out;           // [4,256,1024]

    float* atth   = (float*)d_ws;                            // 1024*512 f32 = 2 MB
    float* weight = (float*)((char*)d_ws + (size_t)BQ * LQ * AH * sizeof(float)); // 1 MB

    // K1: 64*32 = 2048 tiles, 4 waves/block -> 512 blocks
    k_atth<<<512, 128, 0, stream>>>(h, Wh, bh, atth);
    // K2: 4 batches * 32 q-tiles = 128 blocks, 8 waves each
    k_scores<<<128, 256, 0, stream>>>(atth, p, masks, wa, ba, weight);
    // K3: 4*16*64 = 4096 tiles, 8 waves/block -> 512 blocks
    k_out<<<512, 256, 0, stream>>>(weight, feats, outp);
}